// SO3reparameterize_3461743640846
// MI455X (gfx1250) — compile-verified
//
#include <hip/hip_runtime.h>
#include <math.h>

typedef __attribute__((ext_vector_type(2))) float v2f;
typedef __attribute__((ext_vector_type(8))) float v8f;

static constexpr int kD = 512;          // feature dim
static constexpr int kTileM = 16;       // rows per wave tile
static constexpr int kWavesPerBlock = 8;

// ---------------------------------------------------------------------------
// Kernel 1: fused projection  proj[b, 0:9] = x[b,:] @ [W_mu | W_vmu | W_sig]
// via V_WMMA_F32_16X16X4_F32.  One wave32 per 16-row tile.
// ---------------------------------------------------------------------------
__global__ __launch_bounds__(256) void so3_proj_wmma(
    const float* __restrict__ x,
    const float* __restrict__ W_mu,
    const float* __restrict__ W_vmu,
    const float* __restrict__ W_sig,
    float* __restrict__ stash,          // [B][9]  (aliases z sample-0 block)
    int B)
{
    // Combined weight, row-major [k][n], n-stride 16 (cols 9..15 zero). 32 KB.
    __shared__ float lds_w[kD * 16];
    const int tid = threadIdx.x;
    for (int idx = tid; idx < kD * 16; idx += 256) {
        const int kk = idx >> 4;
        const int nn = idx & 15;
        float val = 0.0f;
        if (nn < 3)      val = W_mu [kk * 3 + nn];
        else if (nn < 6) val = W_vmu[kk * 3 + (nn - 3)];
        else if (nn < 9) val = W_sig[kk * 3 + (nn - 6)];
        lds_w[idx] = val;
    }
    __syncthreads();

    const int wave  = tid >> 5;
    const int lane  = tid & 31;
    const int tile  = blockIdx.x * kWavesPerBlock + wave;
    const int tiles = B / kTileM;
    if (tile >= tiles) return;          // uniform per wave -> EXEC all-ones below

    const int m    = lane & 15;         // A row M / B col N
    const int koff = (lane >> 4) * 2;   // K sub-offset for this half-wave

    const float* xrow = x + (size_t)(tile * kTileM + m) * kD;

    v8f acc = {};
    #pragma unroll 4
    for (int k0 = 0; k0 < kD; k0 += 4) {
        // A fragment: 16x4 f32, lane holds x[M, k0+koff .. +1]
        v2f a = *(const v2f*)(xrow + k0 + koff);
        // B fragment: 4x16 f32, lane holds W[k0+koff..+1, N]
        v2f b;
        b.x = lds_w[(k0 + koff)     * 16 + m];
        b.y = lds_w[(k0 + koff + 1) * 16 + m];
        acc = __builtin_amdgcn_wmma_f32_16x16x4_f32(
                  false, a, false, b, (short)0, acc, false, false);
    }

    // C/D layout: VGPR r holds (row M = r + (lane/16)*8, col N = lane&15)
    if (m < 9) {
        const int mbase = tile * kTileM + ((lane >> 4) << 3);
        #pragma unroll
        for (int r = 0; r < 8; ++r)
            stash[(size_t)(mbase + r) * 9 + m] = acc[r];
    }
}

// ---------------------------------------------------------------------------
// Kernel 2: per-row epilogue (bias, softplus, Rodrigues, z = Rmu@Rv, KL)
// ---------------------------------------------------------------------------
__device__ inline void rodrigues3(float w0, float w1, float w2, float R[9]) {
    const float th  = sqrtf(w0 * w0 + w1 * w1 + w2 * w2);
    const float inv = 1.0f / th;
    const float ux = w0 * inv, uy = w1 * inv, uz = w2 * inv;
    const float s = sinf(th), c = cosf(th), cc = 1.0f - c;
    R[0] = c + cc * ux * ux;      R[1] = cc * ux * uy - s * uz; R[2] = cc * ux * uz + s * uy;
    R[3] = cc * uy * ux + s * uz; R[4] = c + cc * uy * uy;      R[5] = cc * uy * uz - s * ux;
    R[6] = cc * uz * ux - s * uy; R[7] = cc * uz * uy + s * ux; R[8] = c + cc * uz * uz;
}

__global__ __launch_bounds__(256) void so3_epilogue(
    const float* __restrict__ stash,    // [B][9] projections (pre-bias)
    const float* __restrict__ b_mu,
    const float* __restrict__ b_vmu,
    const float* __restrict__ b_sig,
    const float* __restrict__ eps,      // [n][B][3]
    const int*   __restrict__ k_ptr,    // wrap count k (device scalar)
    float* __restrict__ z_out,          // [n][B][3][3]
    float* __restrict__ kl_out,         // [B]
    int B, int n)
{
    const int b = blockIdx.x * blockDim.x + threadIdx.x;
    if (b >= B) return;

    const int kang = k_ptr[0];

    float p[9];
    #pragma unroll
    for (int i = 0; i < 9; ++i) p[i] = stash[(size_t)b * 9 + i];

    const float mu0 = p[0] + b_mu[0];
    const float mu1 = p[1] + b_mu[1];
    const float mu2 = p[2] + b_mu[2];
    float vm[3], sg[3];
    float lsig = 0.0f;
    #pragma unroll
    for (int c = 0; c < 3; ++c) {
        vm[c] = p[3 + c] + b_vmu[c];
        const float sp = p[6 + c] + b_sig[c];
        sg[c] = (sp > 20.0f) ? sp : log1pf(expf(sp));   // softplus
        lsig += logf(sg[c]);
    }

    float Rm[9];
    rodrigues3(mu0, mu1, mu2, Rm);

    const float TWO_PI    = 6.28318530717958647692f;
    const float HALF_L2PI3 = 2.75681559961401940f;      // 1.5*log(2*pi)
    const float LOG_8PI2  = 4.36890139479216f;          // log(8*pi^2)
    const float CLAMP     = 0.001f;

    float klacc = 0.0f;
    for (int s = 0; s < n; ++s) {
        const float e0 = eps[((size_t)s * B + b) * 3 + 0];
        const float e1 = eps[((size_t)s * B + b) * 3 + 1];
        const float e2 = eps[((size_t)s * B + b) * 3 + 2];
        const float v0 = vm[0] + e0 * sg[0];
        const float v1 = vm[1] + e1 * sg[1];
        const float v2 = vm[2] + e2 * sg[2];

        float Rv[9];
        rodrigues3(v0, v1, v2, Rv);

        float* zp = z_out + ((size_t)s * B + b) * 9;    // s==0 overwrites stash (already in regs)
        #pragma unroll
        for (int i = 0; i < 3; ++i)
            #pragma unroll
            for (int j = 0; j < 3; ++j)
                zp[i * 3 + j] = Rm[i * 3 + 0] * Rv[0 + j]
                              + Rm[i * 3 + 1] * Rv[3 + j]
                              + Rm[i * 3 + 2] * Rv[6 + j];

        const float th  = sqrtf(v0 * v0 + v1 * v1 + v2 * v2);
        const float it  = 1.0f / th;
        const float u0 = v0 * it, u1 = v1 * it, u2 = v2 * it;

        // streaming logsumexp over 2k+1 wrapped angles
        float mx = -INFINITY, sum = 0.0f;
        for (int t = -kang; t <= kang; ++t) {
            const float thh = th + (float)t * TWO_PI;
            const float z0 = (u0 * thh - vm[0]) / sg[0];
            const float z1 = (u1 * thh - vm[1]) / sg[1];
            const float z2 = (u2 * thh - vm[2]) / sg[2];
            const float lp = -0.5f * (z0 * z0 + z1 * z1 + z2 * z2) - lsig - HALF_L2PI3;
            const float t2  = fmaxf(thh * thh, CLAMP);
            const float den = fmaxf(2.0f - 2.0f * cosf(thh), CLAMP);
            const float val = lp + logf(t2 / den);
            if (val > mx) { sum = sum * expf(mx - val) + 1.0f; mx = val; }
            else          { sum += expf(val - mx); }
        }
        const float logq = mx + logf(sum);
        klacc += logq + LOG_8PI2;
    }
    kl_out[b] = klacc / (float)n;
}

// ---------------------------------------------------------------------------
extern "C" void kernel_launch(void* const* d_in, const int* in_sizes, int n_in,
                              void* d_out, int out_size, void* d_ws, size_t ws_size,
                              hipStream_t stream) {
    const float* x     = (const float*)d_in[0];
    const float* W_mu  = (const float*)d_in[1];
    const float* b_mu  = (const float*)d_in[2];
    const float* W_vmu = (const float*)d_in[3];
    const float* b_vmu = (const float*)d_in[4];
    const float* W_sig = (const float*)d_in[5];
    const float* b_sig = (const float*)d_in[6];
    const float* eps   = (const float*)d_in[7];
    const int*   k_ptr = (const int*)d_in[9];

    const int B = in_sizes[0] / kD;                 // 65536
    const int n = (out_size - B) / (9 * B);         // 4 (out = n*B*9 + B)

    float* out   = (float*)d_out;
    float* stash = out;                             // z sample-0 block reused as scratch
    float* z_out = out;
    float* kl_out = out + (size_t)n * B * 9;

    const int tiles   = B / kTileM;
    const int blocks1 = (tiles + kWavesPerBlock - 1) / kWavesPerBlock;
    so3_proj_wmma<<<blocks1, 256, 0, stream>>>(x, W_mu, W_vmu, W_sig, stash, B);

    const int blocks2 = (B + 255) / 256;
    so3_epilogue<<<blocks2, 256, 0, stream>>>(stash, b_mu, b_vmu, b_sig, eps,
                                              k_ptr, z_out, kl_out, B, n);
}